// online_72344429134228
// MI455X (gfx1250) — compile-verified
//
#include <hip/hip_runtime.h>
#include <math.h>

// Problem constants (from reference)
#define DDIM   2048
#define NROWS  16384
#define ETA    0.01f
// Blocking
#define BLK    64                  // rows per sequential phase
#define CH     128                 // K-chunk staged in LDS
#define NCHUNK (DDIM / CH)         // 16
#define NPHASE (NROWS / BLK)       // 256
#define XPITCH (CH + 4)            // 132 floats: 16B-aligned rows, rotates LDS banks
#define GPITCH (BLK + 1)           // 65 floats: conflict-free column reads in the scan

typedef __attribute__((ext_vector_type(2))) float v2f;
typedef __attribute__((ext_vector_type(8))) float v8f;

__device__ __forceinline__ unsigned lds_off(const void* p) {
  // low 32 bits of a generic (LDS-aperture) address == LDS byte offset
  return (unsigned)(unsigned long long)p;
}

__global__ __launch_bounds__(1024)
void online_sgd_gram_kernel(const float* __restrict__ theta_in,
                            const float* __restrict__ xs,
                            float* __restrict__ theta_out) {
  __shared__ __align__(16) float Xs[BLK][XPITCH];  // staged 64x128 chunk of X
  __shared__ float thetaS[DDIM];                   // live parameters
  __shared__ float G[BLK][GPITCH];                 // Gram block (lower triangle valid)
  __shared__ float bsum[BLK][4];                   // GEMV partials
  __shared__ float bS[BLK];                        // b = X_blk . theta
  __shared__ float cS[BLK];                        // scan coefficients

  const int tid  = (int)threadIdx.x;
  const int lane = tid & 31;
  const int wave = tid >> 5;

  // ---- theta -> LDS ----
  thetaS[tid]        = theta_in[tid];
  thetaS[tid + 1024] = theta_in[tid + 1024];
  __syncthreads();

  // Waves 0..19: lower-triangle Gram tiles (10 tiles x 2 K-halves).
  // Tile t -> (ti,tj), ti >= tj, enumerated row-major over the triangle.
  const int tile = wave >> 1;       // 0..9 (garbage for wave>=20, unused)
  const int kh   = wave & 1;        // K-half: cols [kh*64, kh*64+64)
  int ti = 0, tbase = 0;
  while (tile >= tbase + ti + 1 && ti < 3) { tbase += ti + 1; ++ti; }
  const int tj    = tile - tbase;
  const int m15   = lane & 15;
  const int khalf = (lane >> 4) << 1;   // 0 for lanes 0-15, 2 for lanes 16-31

  // Waves 20..27: GEMV b = X_blk . theta (64 rows x 4 threads)
  const int gvt  = tid - 640;
  const int grow = (gvt >> 2) & 63;
  const int gsub = gvt & 3;

  for (int phase = 0; phase < NPHASE; ++phase) {
    const float* Xblk = xs + (size_t)phase * BLK * DDIM;

    v8f  acc  = {0.f, 0.f, 0.f, 0.f, 0.f, 0.f, 0.f, 0.f};
    float gacc = 0.f;

    for (int ch = 0; ch < NCHUNK; ++ch) {
      __syncthreads();   // previous chunk fully consumed before overwrite
      // ---- async stage: 64x128 f32 -> LDS (2 x b128 per thread) ----
      {
        const int u0 = tid * 2;   // float4 units; 2048 units total
#pragma unroll
        for (int q = 0; q < 2; ++q) {
          const int u  = u0 + q;
          const int r  = u >> 5;        // row 0..63
          const int c4 = u & 31;        // float4 column 0..31
          const float* gp = Xblk + r * DDIM + ch * CH + c4 * 4;
          const unsigned lo = lds_off(&Xs[r][c4 * 4]);
          asm volatile("global_load_async_to_lds_b128 %0, %1, off"
                       :: "v"(lo), "v"(gp) : "memory");
        }
        asm volatile("s_wait_asynccnt 0" ::: "memory");
      }
      __syncthreads();

      if (wave < 20) {
        // ---- Gram tile K-half: acc += A(16x4,f32) x B(4x16,f32), 16 WMMAs ----
        const int k0 = kh * 64;
#pragma unroll 8
        for (int k = k0; k < k0 + 64; k += 4) {
          const int kk = k + khalf;
          v2f a, b;
          a.x = Xs[ti * 16 + m15][kk];
          a.y = Xs[ti * 16 + m15][kk + 1];
          b.x = Xs[tj * 16 + m15][kk];
          b.y = Xs[tj * 16 + m15][kk + 1];
          acc = __builtin_amdgcn_wmma_f32_16x16x4_f32(
                    false, a, false, b, (short)0, acc, false, false);
        }
      } else if (wave < 28) {
        // ---- GEMV partial: b_row += Xs[row][.] * theta[.] (32 cols/chunk) ----
        const float* th = &thetaS[ch * CH + gsub * 32];
        const float* xr = &Xs[grow][gsub * 32];
#pragma unroll
        for (int kk = 0; kk < 32; ++kk) gacc += xr[kk] * th[kk];
      }
    }

    // ---- combine Gram K-half partials into LDS (barrier-ordered) ----
    if (wave < 20 && kh == 0) {
#pragma unroll
      for (int r = 0; r < 8; ++r) {
        const int m = ti * 16 + r + ((lane >> 4) << 3);  // C layout: M=r (+8 hi half)
        G[m][tj * 16 + m15] = acc[r];
      }
    } else if (wave >= 20 && wave < 28) {
      bsum[grow][gsub] = gacc;
    }
    __syncthreads();
    if (wave < 20 && kh == 1) {
#pragma unroll
      for (int r = 0; r < 8; ++r) {
        const int m = ti * 16 + r + ((lane >> 4) << 3);
        G[m][tj * 16 + m15] += acc[r];
      }
    }
    if (tid < BLK) {
      float s = bsum[tid][0] + bsum[tid][1] + bsum[tid][2] + bsum[tid][3];
      bS[tid] = s;
    }
    __syncthreads();

    // ---- sequential 64-step sigmoid scan: rank-1 forward substitution ----
    if (wave == 0) {
      float p0 = bS[lane];        // pred for row 'lane'
      float p1 = bS[lane + 32];   // pred for row 'lane+32'
      for (int j = 0; j < BLK; ++j) {
        const float pj = __shfl((j < 32) ? p0 : p1, j & 31, 32);
        const float cj = ETA / (1.f + expf(pj));   // uniform across lanes
        if (lane == 0) cS[j] = cj;
        if (lane > j)      p0 += cj * G[lane][j];        // stride-65 rows: no conflicts
        if (lane + 32 > j) p1 += cj * G[lane + 32][j];
      }
    }
    __syncthreads();

    // ---- rank-64 update: theta += X_blk^T c (2 columns per thread) ----
    if (phase + 1 < NPHASE)
      __builtin_prefetch(xs + (size_t)(phase + 1) * BLK * DDIM + tid * 16, 0, 0);
#pragma unroll
    for (int h = 0; h < 2; ++h) {
      const int d = tid + h * 1024;
      float t = thetaS[d];
#pragma unroll 8
      for (int j = 0; j < BLK; ++j) t += cS[j] * Xblk[j * DDIM + d];
      thetaS[d] = t;
    }
    __syncthreads();
  }

  theta_out[tid]        = thetaS[tid];
  theta_out[tid + 1024] = thetaS[tid + 1024];
}

extern "C" void kernel_launch(void* const* d_in, const int* in_sizes, int n_in,
                              void* d_out, int out_size, void* d_ws, size_t ws_size,
                              hipStream_t stream) {
  (void)in_sizes; (void)n_in; (void)d_ws; (void)ws_size; (void)out_size;
  const float* theta = (const float*)d_in[0];   // (2048,)
  const float* xs    = (const float*)d_in[1];   // (16384, 2048)
  float* out         = (float*)d_out;           // (2048,)
  online_sgd_gram_kernel<<<1, 1024, 0, stream>>>(theta, xs, out);
}